// EEDGCNEncoder_33036888440992
// MI455X (gfx1250) — compile-verified
//
#include <hip/hip_runtime.h>
#include <hip/hip_bf16.h>
#include <math.h>

#define Bx 4
#define Nx 512
#define Dx 64
#define EDIMx 32
#define Kx 16
#define Lx 2
#define EPSx 1e-5f

typedef __attribute__((ext_vector_type(2))) float v2f;
typedef __attribute__((ext_vector_type(8))) float v8f;

__device__ __forceinline__ float wave_sum32(float x) {
    #pragma unroll
    for (int m = 16; m > 0; m >>= 1) x += __shfl_xor(x, m, 32);
    return x;
}

// ---------------- LayerNorm over H rows (64 elems) : 1 wave per row --------
__global__ void ln_h_kernel(const float* __restrict__ H, const float* __restrict__ g,
                            const float* __restrict__ be, float* __restrict__ out) {
    int wv = threadIdx.x >> 5, lane = threadIdx.x & 31;
    int row = blockIdx.x * 8 + wv;                      // < B*N = 2048
    const float* x = H + row * Dx;
    float x0 = x[lane], x1 = x[lane + 32];
    float m = wave_sum32(x0 + x1) * (1.0f / 64.0f);
    float d0 = x0 - m, d1 = x1 - m;
    float v = wave_sum32(d0 * d0 + d1 * d1) * (1.0f / 64.0f);
    float rs = rsqrtf(v + EPSx);
    out[row * Dx + lane]      = d0 * rs * g[lane]      + be[lane];
    out[row * Dx + lane + 32] = d1 * rs * g[lane + 32] + be[lane + 32];
}

// ---------------- LayerNorm over E rows (32 elems) : 1 wave per row --------
__global__ void ln_e_kernel(const float* __restrict__ E, const float* __restrict__ g,
                            const float* __restrict__ be, float* __restrict__ out) {
    int wv = threadIdx.x >> 5, lane = threadIdx.x & 31;
    long long row = (long long)blockIdx.x * 8 + wv;     // < B*N*N = 1048576
    const float* x = E + row * EDIMx;
    float x0 = x[lane];
    float m = wave_sum32(x0) * (1.0f / 32.0f);
    float d = x0 - m;
    float v = wave_sum32(d * d) * (1.0f / 32.0f);
    out[row * EDIMx + lane] = d * rsqrtf(v + EPSx) * g[lane] + be[lane];
}

// ---------------- squared norms of Hn rows ---------------------------------
__global__ void sq_kernel(const float* __restrict__ Hn, float* __restrict__ sq) {
    int wv = threadIdx.x >> 5, lane = threadIdx.x & 31;
    int row = blockIdx.x * 8 + wv;                      // < 2048
    const float* x = Hn + row * Dx;
    float a = x[lane], b = x[lane + 32];
    float s = wave_sum32(a * a + b * b);
    if (lane == 0) sq[row] = s;
}

// ---------------- d2 = sq_i + sq_j - 2 * x_i . x_j  (WMMA f32) -------------
// one wave per 16x16 tile; K=64 in 16 steps of V_WMMA_F32_16X16X4_F32
__global__ void gram_d2_kernel(const float* __restrict__ Hn, const float* __restrict__ sq,
                               float* __restrict__ d2) {
    __shared__ float Ai[16][64];
    __shared__ float Bj[16][64];
    int bid = blockIdx.x;                               // ((b*32)+it)*32 + jt
    int jt = bid & 31, it = (bid >> 5) & 31, b = bid >> 10;
    int i0 = it * 16, j0 = jt * 16;
    int lane = threadIdx.x;
    const float* xb = Hn + b * Nx * Dx;
    #pragma unroll
    for (int q = 0; q < 8; ++q) {
        int vi = q * 32 + lane;                         // float4 index (0..255)
        int r = vi >> 4, c = (vi & 15) * 4;
        *(float4*)&Ai[r][c] = *(const float4*)&xb[(i0 + r) * Dx + c];
        *(float4*)&Bj[r][c] = *(const float4*)&xb[(j0 + r) * Dx + c];
    }
    __syncthreads();
    int rr = lane & 15, hi = lane >> 4;
    v8f acc = {};
    #pragma unroll
    for (int kk = 0; kk < 64; kk += 4) {
        int ko = kk + 2 * hi;
        v2f a  = { Ai[rr][ko], Ai[rr][ko + 1] };
        v2f bb = { Bj[rr][ko], Bj[rr][ko + 1] };
        acc = __builtin_amdgcn_wmma_f32_16x16x4_f32(false, a, false, bb, (short)0, acc, false, false);
    }
    float sj = sq[b * Nx + j0 + rr];
    #pragma unroll
    for (int vv = 0; vv < 8; ++vv) {
        int mi = vv + 8 * hi;
        d2[(b * Nx + i0 + mi) * Nx + j0 + rr] = sq[b * Nx + i0 + mi] + sj - 2.0f * acc[vv];
    }
}

// ---------------- top-16 smallest per row (register insertion chain) -------
__global__ void topk_kernel(const float* __restrict__ d2, int* __restrict__ idx) {
    int row = blockIdx.x * blockDim.x + threadIdx.x;    // < 2048
    const float* r = d2 + row * Nx;
    float best[16]; int bi[16];
    #pragma unroll
    for (int t = 0; t < 16; ++t) { best[t] = 3.4e38f; bi[t] = 0; }
    for (int j = 0; j < Nx; ++j) {
        float cv = r[j]; int ci = j;
        if (cv < best[15]) {
            #pragma unroll
            for (int t = 0; t < 16; ++t) {
                if (cv < best[t]) {
                    float tf = best[t]; best[t] = cv; cv = tf;
                    int   ti = bi[t];   bi[t]   = ci; ci = ti;
                }
            }
        }
    }
    #pragma unroll
    for (int t = 0; t < 16; ++t) idx[row * Kx + t] = bi[t];
}

// ---------------- h = comb @ mlp_w  (M=32768, K=128, N=64) WMMA f32 --------
// one wave per 16-row M tile; comb built in LDS from gathers
__global__ void edgemlp_kernel(const float* __restrict__ Hn, const int* __restrict__ idx,
                               const float* __restrict__ mw, float* __restrict__ h) {
    __shared__ float comb[16][128];
    int tile = blockIdx.x;                              // < 2048
    int lane = threadIdx.x;
    int base = tile * 16;
    #pragma unroll
    for (int m = 0; m < 16; ++m) {
        int flat = base + m;                            // = (b*N+n)*K + k
        int n = (flat >> 4) & (Nx - 1);
        int b = flat >> 13;
        const float* cen = Hn + (b * Nx + n) * Dx;
        const float* nb  = Hn + (b * Nx + idx[flat]) * Dx;
        float c0 = cen[lane], c1 = cen[lane + 32];
        comb[m][lane]      = c0;
        comb[m][lane + 32] = c1;
        comb[m][64 + lane]      = nb[lane]      - c0;
        comb[m][96 + lane]      = nb[lane + 32] - c1;
    }
    __syncthreads();
    int rr = lane & 15, hi = lane >> 4;
    #pragma unroll
    for (int n0 = 0; n0 < 64; n0 += 16) {
        v8f acc = {};
        for (int kk = 0; kk < 128; kk += 4) {
            int ko = kk + 2 * hi;
            v2f a  = { comb[rr][ko], comb[rr][ko + 1] };
            v2f bb = { mw[ko * Dx + n0 + rr], mw[(ko + 1) * Dx + n0 + rr] };
            acc = __builtin_amdgcn_wmma_f32_16x16x4_f32(false, a, false, bb, (short)0, acc, false, false);
        }
        #pragma unroll
        for (int vv = 0; vv < 8; ++vv) {
            int m = vv + 8 * hi;
            h[(base + m) * Dx + n0 + rr] = acc[vv];
        }
    }
}

// ---------------- BN statistics over 32768 rows, per channel ---------------
__global__ void bnstats_kernel(const float* __restrict__ h, const float* __restrict__ gamma,
                               const float* __restrict__ beta, float* __restrict__ scale,
                               float* __restrict__ shift) {
    const int ROWS = Bx * Nx * Kx;                      // 32768
    int d = blockIdx.x, tid = threadIdx.x;
    float s = 0.f, ss = 0.f;
    for (int r = tid; r < ROWS; r += 256) {
        float x = h[r * Dx + d];
        s += x; ss += x * x;
    }
    __shared__ float S[256], SS[256];
    S[tid] = s; SS[tid] = ss; __syncthreads();
    for (int o = 128; o > 0; o >>= 1) {
        if (tid < o) { S[tid] += S[tid + o]; SS[tid] += SS[tid + o]; }
        __syncthreads();
    }
    if (tid == 0) {
        float mu  = S[0]  * (1.0f / ROWS);
        float var = SS[0] * (1.0f / ROWS) - mu * mu;
        float sc  = gamma[d] * rsqrtf(var + EPSx);
        scale[d] = sc;
        shift[d] = beta[d] - mu * sc;
    }
}

// ---------------- BN + exact GELU + mean over K  -> n_out (=out_H) ---------
__global__ void bngelumean_kernel(const float* __restrict__ h, const float* __restrict__ scale,
                                  const float* __restrict__ shift, float* __restrict__ nout) {
    int blk = blockIdx.x;                               // < 2048 (= b*N+n)
    int d = threadIdx.x;
    float sc = scale[d], sh = shift[d];
    float acc = 0.f;
    #pragma unroll
    for (int k = 0; k < Kx; ++k) {
        float x = h[(blk * Kx + k) * Dx + d] * sc + sh;
        acc += 0.5f * x * (1.0f + erff(x * 0.70710678118654752f));
    }
    nout[blk * Dx + d] = acc * (1.0f / Kx);
}

// ---------------- g1 = n_out@w_n1, g2 = n_out@w_n2 -------------------------
__global__ void g12_kernel(const float* __restrict__ nout, const float* __restrict__ hw,
                           float* __restrict__ g1, float* __restrict__ g2) {
    int blk = blockIdx.x, t = threadIdx.x;
    int f = t & 31; bool second = t >= 32;
    const float* w = hw + (second ? (EDIMx + Dx) : EDIMx) * EDIMx;
    const float* x = nout + blk * Dx;
    float acc = 0.f;
    #pragma unroll 8
    for (int dd = 0; dd < Dx; ++dd) acc += x[dd] * w[dd * EDIMx + f];
    (second ? g2 : g1)[blk * EDIMx + f] = acc;
}

// ---------------- e_out = En@w_e + g1[b,j] + g2[b,i] + hb  (in place) ------
// 8 waves/block, each wave one 16-row (j) tile of 32 channels, WMMA f32.
// Edge tile staged into LDS with GLOBAL_LOAD_ASYNC_TO_LDS_B128 (ASYNCcnt).
__global__ void eout_kernel(float* __restrict__ E, const float* __restrict__ hw,
                            const float* __restrict__ hb, const float* __restrict__ g1,
                            const float* __restrict__ g2) {
    __shared__ float We[32][32];
    __shared__ float Et[8][16][32];
    int tid = threadIdx.x, lane = tid & 31, wv = tid >> 5;
    for (int q = tid; q < 1024; q += 256) We[q >> 5][q & 31] = hw[q]; // w_e rows 0..31
    int bid = blockIdx.x;                               // ((b*N)+i)*4 + jblk
    int jblk = bid & 3, i = (bid >> 2) & (Nx - 1), b = bid >> 11;
    int j0 = jblk * 128 + wv * 16;
    float* ebase = E + ((long long)(b * Nx + i) * Nx + j0) * EDIMx;   // 16 rows x 32, contiguous

    // Async copy 2KB tile (global -> LDS) without the VGPR round-trip.
    // Low 32 bits of a generic pointer to __shared__ are the LDS byte offset
    // (flat->LDS aperture truncation, ISA 10.2).
    unsigned ldsbase = (unsigned)(unsigned long long)&Et[wv][0][0];
    #pragma unroll
    for (int q = 0; q < 4; ++q) {
        int vi = q * 32 + lane;                         // float4 index (0..127)
        unsigned loff = ldsbase + (unsigned)vi * 16u;
        const float4* ga = (const float4*)ebase + vi;
        asm volatile("global_load_async_to_lds_b128 %0, %1, off"
                     :: "v"(loff), "v"(ga)
                     : "memory");
    }
    asm volatile("s_wait_asynccnt 0x0" ::: "memory");
    __syncthreads();

    int rr = lane & 15, hi = lane >> 4;
    #pragma unroll
    for (int n0 = 0; n0 < 32; n0 += 16) {
        v8f acc = {};
        #pragma unroll
        for (int kk = 0; kk < 32; kk += 4) {
            int ko = kk + 2 * hi;
            v2f a  = { Et[wv][rr][ko], Et[wv][rr][ko + 1] };
            v2f bb = { We[ko][n0 + rr], We[ko + 1][n0 + rr] };
            acc = __builtin_amdgcn_wmma_f32_16x16x4_f32(false, a, false, bb, (short)0, acc, false, false);
        }
        int n = n0 + rr;
        float addc = g2[(b * Nx + i) * EDIMx + n] + hb[n];
        #pragma unroll
        for (int vv = 0; vv < 8; ++vv) {
            int j = j0 + vv + 8 * hi;
            ebase[(vv + 8 * hi) * EDIMx + n] = acc[vv] + g1[(b * Nx + j) * EDIMx + n] + addc;
        }
    }
}

extern "C" void kernel_launch(void* const* d_in, const int* in_sizes, int n_in,
                              void* d_out, int out_size, void* d_ws, size_t ws_size,
                              hipStream_t stream) {
    const float* H      = (const float*)d_in[0];
    const float* E      = (const float*)d_in[1];
    const float* mlp_w  = (const float*)d_in[2];   // (L,128,64)
    const float* bn_g   = (const float*)d_in[3];   // (L,64)
    const float* bn_b   = (const float*)d_in[4];
    const float* hw_w   = (const float*)d_in[5];   // (L,160,32)
    const float* hw_b   = (const float*)d_in[6];   // (L,32)
    const float* lnH_g  = (const float*)d_in[7];
    const float* lnH_b  = (const float*)d_in[8];
    const float* lnE_g  = (const float*)d_in[9];
    const float* lnE_b  = (const float*)d_in[10];

    float* outH = (float*)d_out;                       // B*N*D
    float* outE = outH + Bx * Nx * Dx;                 // B*N*N*EDIM

    char* ws = (char*)d_ws;
    float* sq  = (float*)ws;                  ws += ((Bx * Nx * 4 + 255) & ~255);
    float* d2  = (float*)ws;                  ws += ((size_t)Bx * Nx * Nx * 4 + 255) & ~(size_t)255;
    int*   idx = (int*)ws;                    ws += ((Bx * Nx * Kx * 4 + 255) & ~255);
    float* h   = (float*)ws;                  ws += ((size_t)Bx * Nx * Kx * Dx * 4 + 255) & ~(size_t)255;
    float* bsc = (float*)ws;                  ws += 256;
    float* bsh = (float*)ws;                  ws += 256;
    float* g1  = (float*)ws;                  ws += ((Bx * Nx * EDIMx * 4 + 255) & ~255);
    float* g2  = (float*)ws;                  ws += ((Bx * Nx * EDIMx * 4 + 255) & ~255);

    ln_h_kernel<<<Bx * Nx / 8, 256, 0, stream>>>(H, lnH_g, lnH_b, outH);
    ln_e_kernel<<<Bx * Nx * Nx / 8, 256, 0, stream>>>(E, lnE_g, lnE_b, outE);

    for (int l = 0; l < Lx; ++l) {
        const float* mw = mlp_w + l * 2 * Dx * Dx;              // 8192
        const float* hw = hw_w  + l * (EDIMx + 2 * Dx) * EDIMx; // 5120
        sq_kernel<<<Bx * Nx / 8, 256, 0, stream>>>(outH, sq);
        gram_d2_kernel<<<Bx * (Nx / 16) * (Nx / 16), 32, 0, stream>>>(outH, sq, d2);
        topk_kernel<<<Bx * Nx / 256, 256, 0, stream>>>(d2, idx);
        edgemlp_kernel<<<Bx * Nx * Kx / 16, 32, 0, stream>>>(outH, idx, mw, h);
        bnstats_kernel<<<Dx, 256, 0, stream>>>(h, bn_g + l * Dx, bn_b + l * Dx, bsc, bsh);
        bngelumean_kernel<<<Bx * Nx, Dx, 0, stream>>>(h, bsc, bsh, outH);
        g12_kernel<<<Bx * Nx, 64, 0, stream>>>(outH, hw, g1, g2);
        eout_kernel<<<Bx * Nx * (Nx / 128), 256, 0, stream>>>(outE, hw, hw_b + l * EDIMx, g1, g2);
    }
    (void)in_sizes; (void)n_in; (void)out_size; (void)ws_size;
}